// CausalLinearAttentionPy_46325517254673
// MI455X (gfx1250) — compile-verified
//
#include <hip/hip_runtime.h>

// Causal linear attention (fp32) on gfx1250 via V_WMMA_F32_16X16X4_F32 +
// GLOBAL_LOAD_ASYNC_TO_LDS_B128 (CDNA5 async global->LDS path).
// 3 passes: (1) per-chunk K^T V + key sums, (2) exclusive prefix over chunks,
// (3) per-chunk output: tril(q k^T) v + q S, with the 1/(q . cumsum k + eps)
// normalizer folded in as an extra WMMA output tile (B = ones / key-sums).

#define N_  4
#define L_  8192
#define H_  8
#define D_  64
#define T_  128       // chunk length
#define C_  64        // chunks = L_/T_
#define NH_ 32        // N_*H_
#define P_  68        // padded LDS row stride (floats): 272B = 16B-aligned,
                      // bank = (4*row + col) % 64 -> conflict-free patterns

typedef float v2f __attribute__((ext_vector_type(2)));
typedef float v8f __attribute__((ext_vector_type(8)));

__device__ __forceinline__ v8f wmma4(v2f a, v2f b, v8f c) {
  // D = A(16x4) * B(4x16) + C(16x16), all fp32
  return __builtin_amdgcn_wmma_f32_16x16x4_f32(false, a, false, b, (short)0, c,
                                               false, false);
}

__device__ __forceinline__ float fmap(float x) {  // elu(x)+1
  return x > 0.f ? x + 1.f : __expf(x);
}

// Async global->LDS copy, 16 bytes per lane. VDST VGPR carries the LDS byte
// offset, VADDR the 64-bit global address (GV mode). Tracked on ASYNCcnt.
__device__ __forceinline__ void async_b128(unsigned lds_off,
                                           unsigned long long gaddr) {
  asm volatile("global_load_async_to_lds_b128 %0, %1, off"
               :: "v"(lds_off), "v"(gaddr) : "memory");
}
__device__ __forceinline__ void wait_async() {
  asm volatile("s_wait_asynccnt 0" ::: "memory");
}
__device__ __forceinline__ unsigned lds_off(const float* p) {
  return (unsigned)(size_t)p;  // generic LDS addr: low 32 bits = LDS offset
}

// ---------------- Pass 1: per-chunk M_c = k^T v (64x64) and ksum_c ----------
__global__ __launch_bounds__(256) void k_chunksums(
    const float* __restrict__ Kg, const float* __restrict__ Vg,
    float* __restrict__ Mbuf, float* __restrict__ Ksum) {
  extern __shared__ float sm[];
  float* sk = sm;             // [128][P_] feature-mapped keys
  float* sv = sm + T_ * P_;   // [128][P_] values
  const int blk = blockIdx.x;           // nh*C_ + c
  const int c   = blk & (C_ - 1);
  const int nh  = blk >> 6;
  const int n   = nh >> 3, h = nh & (H_ - 1);
  const long t0 = (long)c * T_;
  const int tid = threadIdx.x;

  // values: async DMA into LDS (raw copy, no VGPR involvement)
#pragma unroll
  for (int it = 0; it < 8; ++it) {
    int idx = tid + 256 * it;            // 0..2047 16B-chunks
    int r = idx >> 4, j4 = (idx & 15) * 4;
    unsigned long long g = (unsigned long long)(
        Vg + (((size_t)n * L_ + (t0 + r)) * H_ + h) * D_ + j4);
    async_b128(lds_off(&sv[r * P_ + j4]), g);
  }
  // keys: need elu(x)+1, go through VGPRs
  for (int i = tid; i < T_ * D_; i += 256) {
    int r = i >> 6, d = i & 63;
    size_t g = (((size_t)n * L_ + (t0 + r)) * H_ + h) * D_ + d;
    sk[r * P_ + d] = fmap(Kg[g]);
  }
  wait_async();
  __syncthreads();

  const int wave = tid >> 5, lane = tid & 31;
  const int col = lane & 15, half = lane >> 4, kb = half * 2;
  float* mb = Mbuf + (size_t)blk * 4096;

  for (int t = wave * 2; t < wave * 2 + 2; ++t) {  // 16 tiles over 8 waves
    int ti = t >> 2, tj = t & 3;
    v8f acc = {}, acc2 = {};
    for (int s = 0; s < 32; s += 2) {              // K-dim = 128, dual chains
      int k0 = 4 * s + kb;
      v2f a0 = { sk[k0 * P_ + 16 * ti + col], sk[(k0 + 1) * P_ + 16 * ti + col] };
      v2f b0 = { sv[k0 * P_ + 16 * tj + col], sv[(k0 + 1) * P_ + 16 * tj + col] };
      acc = wmma4(a0, b0, acc);
      int k1 = k0 + 4;
      v2f a1 = { sk[k1 * P_ + 16 * ti + col], sk[(k1 + 1) * P_ + 16 * ti + col] };
      v2f b1 = { sv[k1 * P_ + 16 * tj + col], sv[(k1 + 1) * P_ + 16 * tj + col] };
      acc2 = wmma4(a1, b1, acc2);
    }
#pragma unroll
    for (int r = 0; r < 8; ++r)
      mb[(16 * ti + half * 8 + r) * 64 + 16 * tj + col] = acc[r] + acc2[r];
  }
  if (tid < 64) {  // per-d key sum for the normalizer
    float s = 0.f;
    for (int r = 0; r < T_; ++r) s += sk[r * P_ + tid];
    Ksum[(size_t)blk * 64 + tid] = s;
  }
}

// ---------------- Pass 2: exclusive prefix over chunks ----------------------
__global__ __launch_bounds__(256) void k_prefix(float* __restrict__ Mbuf,
                                                float* __restrict__ Ksum) {
  const int nh = blockIdx.x, tid = threadIdx.x;
  float run[16];
#pragma unroll
  for (int i = 0; i < 16; ++i) run[i] = 0.f;
  size_t base = (size_t)nh * C_ * 4096;
  for (int c = 0; c < C_; ++c) {
    size_t b = base + (size_t)c * 4096;
#pragma unroll
    for (int i = 0; i < 16; ++i) {
      size_t idx = b + tid + 256 * i;
      float v = Mbuf[idx];
      Mbuf[idx] = run[i];
      run[i] += v;
    }
  }
  if (tid < 64) {
    float r = 0.f;
    size_t kb = (size_t)nh * C_ * 64;
    for (int c = 0; c < C_; ++c) {
      size_t idx = kb + (size_t)c * 64 + tid;
      float v = Ksum[idx];
      Ksum[idx] = r;
      r += v;
    }
  }
}

// ---------------- Pass 3: out = (tril(q k^T) v + q S) * Z -------------------
__global__ __launch_bounds__(256) void k_main(
    const float* __restrict__ Qg, const float* __restrict__ Kg,
    const float* __restrict__ Vg, const float* __restrict__ Mbuf,
    const float* __restrict__ Ksum, float* __restrict__ Out) {
  extern __shared__ float sm[];
  float* sk    = sm;                 // [128][P_] feature-mapped keys
  float* sv    = sk + T_ * P_;       // [128][P_] values           (async DMA)
  float* ss    = sv + T_ * P_;       // [64][P_]  prefix state S   (async DMA)
  float* sks   = ss + 64 * P_;       // [64]      prefix key sums
  float* stile = sks + 64;           // 8 waves x 2 x [16][17] attn tiles

  const int blk = blockIdx.x;
  const int c = blk & (C_ - 1);
  const int nh = blk >> 6;
  const int n = nh >> 3, h = nh & (H_ - 1);
  const long t0 = (long)c * T_;
  const int tid = threadIdx.x;

  // v and S: async DMA to LDS (raw copies)
#pragma unroll
  for (int it = 0; it < 8; ++it) {
    int idx = tid + 256 * it;
    int r = idx >> 4, j4 = (idx & 15) * 4;
    unsigned long long g = (unsigned long long)(
        Vg + (((size_t)n * L_ + (t0 + r)) * H_ + h) * D_ + j4);
    async_b128(lds_off(&sv[r * P_ + j4]), g);
  }
#pragma unroll
  for (int it = 0; it < 4; ++it) {
    int idx = tid + 256 * it;            // 0..1023 16B-chunks of S
    int r = idx >> 4, j4 = (idx & 15) * 4;
    async_b128(lds_off(&ss[r * P_ + j4]),
               (unsigned long long)(Mbuf + (size_t)blk * 4096 + idx * 4));
  }
  // keys: elu(x)+1 through VGPRs
  for (int i = tid; i < T_ * D_; i += 256) {
    int r = i >> 6, d = i & 63;
    size_t g = (((size_t)n * L_ + (t0 + r)) * H_ + h) * D_ + d;
    sk[r * P_ + d] = fmap(Kg[g]);
  }
  if (tid < 64) sks[tid] = Ksum[(size_t)blk * 64 + tid];
  wait_async();
  __syncthreads();

  const int wave = tid >> 5, lane = tid & 31;
  const int row = lane & 15, half = lane >> 4, kb = half * 2;

  // q A-fragments for this wave's 16 rows (feature-mapped, from global)
  v2f qa[16];
  {
    size_t gq = (((size_t)n * L_ + (t0 + 16 * wave + row)) * H_ + h) * D_;
#pragma unroll
    for (int ks = 0; ks < 16; ++ks) {
      qa[ks][0] = fmap(Qg[gq + 4 * ks + kb]);
      qa[ks][1] = fmap(Qg[gq + 4 * ks + kb + 1]);
    }
  }

  v8f acc0 = {}, acc1 = {}, acc2 = {}, acc3 = {}, accD = {};

  // Intra-chunk: attn tiles jj <= wave (causality), fused with attn*v
  for (int jj = 0; jj <= wave; ++jj) {
    v8f atA = {}, atB = {};  // dual accumulator chains
#pragma unroll
    for (int ks = 0; ks < 16; ks += 2) {  // K-dim = D = 64
      int d0 = 4 * ks + kb;
      v2f b0 = { sk[(16 * jj + row) * P_ + d0],
                 sk[(16 * jj + row) * P_ + d0 + 1] };
      atA = wmma4(qa[ks], b0, atA);
      int d1 = d0 + 4;
      v2f b1 = { sk[(16 * jj + row) * P_ + d1],
                 sk[(16 * jj + row) * P_ + d1 + 1] };
      atB = wmma4(qa[ks + 1], b1, atB);
    }
    v8f at;
#pragma unroll
    for (int r = 0; r < 8; ++r) at[r] = atA[r] + atB[r];
    if (jj == wave) {  // tril mask on diagonal tile: keep if m >= ncol
#pragma unroll
      for (int r = 0; r < 8; ++r)
        if (half * 8 + r < row) at[r] = 0.f;
    }
    float* tb = stile + (wave * 2 + (jj & 1)) * 16 * 17;  // double-buffered
#pragma unroll
    for (int r = 0; r < 8; ++r)  // D-layout -> LDS for A-fragment reuse
      tb[(half * 8 + r) * 17 + row] = at[r];

#pragma unroll
    for (int s = 0; s < 4; ++s) {  // attn_tile (A) x v rows 16jj..16jj+15 (B)
      v2f a = { tb[row * 17 + 4 * s + kb], tb[row * 17 + 4 * s + kb + 1] };
      int r0 = 16 * jj + 4 * s + kb;
      v2f bd = { 1.f, 1.f };  // denominator tile: attn @ ones = row sums
      accD = wmma4(a, bd, accD);
      v2f b0 = { sv[r0 * P_ + row],      sv[(r0 + 1) * P_ + row] };
      acc0 = wmma4(a, b0, acc0);
      v2f b1 = { sv[r0 * P_ + 16 + row], sv[(r0 + 1) * P_ + 16 + row] };
      acc1 = wmma4(a, b1, acc1);
      v2f b2 = { sv[r0 * P_ + 32 + row], sv[(r0 + 1) * P_ + 32 + row] };
      acc2 = wmma4(a, b2, acc2);
      v2f b3 = { sv[r0 * P_ + 48 + row], sv[(r0 + 1) * P_ + 48 + row] };
      acc3 = wmma4(a, b3, acc3);
    }
  }

  // Inter-chunk: q @ S (+ q . prefix-ksum into the denominator tile)
#pragma unroll
  for (int ks = 0; ks < 16; ++ks) {
    int d0 = 4 * ks + kb;
    v2f bd = { sks[d0], sks[d0 + 1] };
    accD = wmma4(qa[ks], bd, accD);
    v2f b0 = { ss[d0 * P_ + row],      ss[(d0 + 1) * P_ + row] };
    acc0 = wmma4(qa[ks], b0, acc0);
    v2f b1 = { ss[d0 * P_ + 16 + row], ss[(d0 + 1) * P_ + 16 + row] };
    acc1 = wmma4(qa[ks], b1, acc1);
    v2f b2 = { ss[d0 * P_ + 32 + row], ss[(d0 + 1) * P_ + 32 + row] };
    acc2 = wmma4(qa[ks], b2, acc2);
    v2f b3 = { ss[d0 * P_ + 48 + row], ss[(d0 + 1) * P_ + 48 + row] };
    acc3 = wmma4(qa[ks], b3, acc3);
  }

  // Z normalize + store (denominator tile has identical columns, lane-matched)
#pragma unroll
  for (int r = 0; r < 8; ++r) {
    int m = half * 8 + r;
    size_t go = (((size_t)n * L_ + (t0 + 16 * wave + m)) * H_ + h) * D_;
    float z = 1.f / (accD[r] + 1e-6f);
    Out[go + row]      = acc0[r] * z;
    Out[go + 16 + row] = acc1[r] * z;
    Out[go + 32 + row] = acc2[r] * z;
    Out[go + 48 + row] = acc3[r] * z;
  }
}

extern "C" void kernel_launch(void* const* d_in, const int* in_sizes, int n_in,
                              void* d_out, int out_size, void* d_ws,
                              size_t ws_size, hipStream_t stream) {
  const float* Qg = (const float*)d_in[0];
  const float* Kg = (const float*)d_in[1];
  const float* Vg = (const float*)d_in[2];
  float* Out = (float*)d_out;
  float* Mbuf = (float*)d_ws;                       // [NH, C, 64, 64] fp32
  float* Ksum = Mbuf + (size_t)NH_ * C_ * 64 * 64;  // [NH, C, 64]     fp32

  const size_t smem1 = (size_t)(2 * T_ * P_) * sizeof(float);
  const size_t smem3 =
      (size_t)(2 * T_ * P_ + 64 * P_ + 64 + 2 * 8 * 16 * 17) * sizeof(float);

  k_chunksums<<<dim3(NH_ * C_), dim3(256), smem1, stream>>>(Kg, Vg, Mbuf, Ksum);
  k_prefix<<<dim3(NH_), dim3(256), 0, stream>>>(Mbuf, Ksum);
  k_main<<<dim3(NH_ * C_), dim3(256), smem3, stream>>>(Qg, Kg, Vg, Mbuf, Ksum,
                                                       Out);
}